// MultiHeadGATLayer_24575802867851
// MI455X (gfx1250) — compile-verified
//
#include <hip/hip_runtime.h>
#include <hip/hip_bf16.h>
#include <float.h>

#define N_NODES 50000
#define E_EDGES 800000
#define F_IN    128
#define F_OUT   16
#define H_HEADS 8
#define F_TOT   (F_OUT * H_HEADS)   // 128
#define SLOPE   0.2f
#define XS_STRIDE 132               // LDS row pad: bank stride 4 -> conflict-free

typedef __attribute__((ext_vector_type(2))) float v2f;
typedef __attribute__((ext_vector_type(8))) float v8f;

// ---------------------------------------------------------------------------
// Kernel 1: projection GEMM h = X @ W^T + b  using V_WMMA_F32_16X16X4_F32.
// One workgroup (8 waves) per 16-node tile; wave w computes the 16x16 output
// tile for head w (cols o0 = w*16). K=128 -> 32 chained WMMA ops per wave.
// ---------------------------------------------------------------------------
__global__ __launch_bounds__(256) void gat_project_wmma(
    const float* __restrict__ X, const float* __restrict__ W,
    const float* __restrict__ bias, float* __restrict__ h)
{
    __shared__ float xs[16 * XS_STRIDE];

    const int node_base = blockIdx.x * 16;
    const int t = threadIdx.x;

    // Cooperative load of the 16x128 fp32 X tile (2048 floats, 8 per thread).
    {
        int flat = t * 8;
        int row  = flat >> 7;        // /128
        int col  = flat & 127;
        const float4* src = (const float4*)(X + (size_t)(node_base + row) * F_IN + col);
        float4 p0 = src[0];
        float4 p1 = src[1];
        float4* dst = (float4*)(&xs[row * XS_STRIDE + col]);
        dst[0] = p0;
        dst[1] = p1;
    }
    __syncthreads();

    const int lane  = t & 31;
    const int nn    = lane & 15;     // A: row m | B: col n (both = lane % 16)
    const int khalf = lane >> 4;     // K sub-pair select (ISA 16x4 f32 layout)
    const int o0    = (t >> 5) * 16; // wave id * 16 = output-column base (head)

    const float* xrow = &xs[nn * XS_STRIDE];              // X row m
    const float* wrow = W + (size_t)(o0 + nn) * F_IN;     // W row for col n

    v8f acc = {};
#pragma unroll
    for (int k0 = 0; k0 < F_IN; k0 += 4) {
        const int kk = k0 + khalf * 2;
        v2f av, bv;
        av[0] = xrow[kk];
        av[1] = xrow[kk + 1];
        bv[0] = wrow[kk];
        bv[1] = wrow[kk + 1];
        // D = A(16x4, f32) x B(4x16, f32) + C(16x16, f32)
        acc = __builtin_amdgcn_wmma_f32_16x16x4_f32(
            false, av, false, bv, (short)0, acc, false, false);
    }

    const float bb = bias[o0 + nn];  // col n fixed per lane -> one bias value
#pragma unroll
    for (int r = 0; r < 8; ++r) {
        const int mrow = r + khalf * 8;  // C/D layout: lanes 16-31 hold M=8..15
        h[(size_t)(node_base + mrow) * F_TOT + o0 + nn] = acc[r] + bb;
    }
}

// ---------------------------------------------------------------------------
// Kernel 2: per-node attention scalars s_t = h . a[:16], s_s = h . a[16:]
// ---------------------------------------------------------------------------
__global__ void gat_node_scores(const float* __restrict__ h,
                                const float* __restrict__ a,
                                float* __restrict__ s_t, float* __restrict__ s_s)
{
    int idx = blockIdx.x * blockDim.x + threadIdx.x;
    if (idx >= N_NODES * H_HEADS) return;
    const int n  = idx >> 3;
    const int hd = idx & 7;
    const float* hp = h + (size_t)n * F_TOT + hd * F_OUT;
    float st = 0.f, ss = 0.f;
#pragma unroll
    for (int f = 0; f < F_OUT; ++f) {
        const float v = hp[f];
        st += v * a[f];
        ss += v * a[F_OUT + f];
    }
    s_t[idx] = st;
    s_s[idx] = ss;
}

// ---------------------------------------------------------------------------
// Kernel 3: init accumulators (harness poisons ws with 0xAA each run)
// ---------------------------------------------------------------------------
__global__ void gat_init(float* __restrict__ acc, float* __restrict__ segmax,
                         float* __restrict__ segsum, float* __restrict__ deg)
{
    int idx = blockIdx.x * blockDim.x + threadIdx.x;
    if (idx < N_NODES * F_TOT) acc[idx] = 0.f;
    if (idx < N_NODES * H_HEADS) { segmax[idx] = -FLT_MAX; segsum[idx] = 0.f; }
    if (idx < N_NODES) deg[idx] = 0.f;
}

// Sign-aware float atomic max via int/uint monotonic bit tricks.
__device__ __forceinline__ void atomicMaxFloat(float* addr, float value)
{
    if (value >= 0.f)
        atomicMax((int*)addr, __float_as_int(value));
    else
        atomicMin((unsigned int*)addr, __float_as_uint(value));
}

__device__ __forceinline__ float leaky(float x)
{
    return x > 0.f ? x : SLOPE * x;
}

// ---------------------------------------------------------------------------
// Kernel 4: per-edge segment max (+ in-degree count for the skip connection)
// ---------------------------------------------------------------------------
__global__ void gat_edge_max(const int* __restrict__ ei,
                             const float* __restrict__ s_t,
                             const float* __restrict__ s_s,
                             float* __restrict__ segmax, float* __restrict__ deg)
{
    int e = blockIdx.x * blockDim.x + threadIdx.x;
    if (e >= E_EDGES) return;
    const int src = ei[e];
    const int tgt = ei[E_EDGES + e];
    const float* st = s_t + (size_t)tgt * H_HEADS;
    const float* ss = s_s + (size_t)src * H_HEADS;
#pragma unroll
    for (int hd = 0; hd < H_HEADS; ++hd) {
        atomicMaxFloat(&segmax[(size_t)tgt * H_HEADS + hd], leaky(st[hd] + ss[hd]));
    }
    atomicAdd(&deg[tgt], 1.0f);
}

// ---------------------------------------------------------------------------
// Kernel 5: per-edge segment sum of exp(e - max)
// ---------------------------------------------------------------------------
__global__ void gat_edge_sum(const int* __restrict__ ei,
                             const float* __restrict__ s_t,
                             const float* __restrict__ s_s,
                             const float* __restrict__ segmax,
                             float* __restrict__ segsum)
{
    int e = blockIdx.x * blockDim.x + threadIdx.x;
    if (e >= E_EDGES) return;
    const int src = ei[e];
    const int tgt = ei[E_EDGES + e];
    const float* st = s_t + (size_t)tgt * H_HEADS;
    const float* ss = s_s + (size_t)src * H_HEADS;
    const float* mx = segmax + (size_t)tgt * H_HEADS;
#pragma unroll
    for (int hd = 0; hd < H_HEADS; ++hd) {
        const float ex = __expf(leaky(st[hd] + ss[hd]) - mx[hd]);
        atomicAdd(&segsum[(size_t)tgt * H_HEADS + hd], ex);
    }
}

// ---------------------------------------------------------------------------
// Kernel 6: message scatter  acc[tgt, f] += alpha[tgt_edge, f/16] * h[src, f]
// 256 threads = 2 edges/block, 128 lanes per edge (one per feature column).
// ---------------------------------------------------------------------------
__global__ __launch_bounds__(256) void gat_edge_scatter(
    const int* __restrict__ ei, const float* __restrict__ h,
    const float* __restrict__ s_t, const float* __restrict__ s_s,
    const float* __restrict__ segmax, const float* __restrict__ segsum,
    float* __restrict__ acc)
{
    const int le = threadIdx.x >> 7;       // local edge 0/1
    const int f  = threadIdx.x & 127;      // feature column
    const int hd = f >> 4;                 // head
    const long e = (long)blockIdx.x * 2 + le;
    if (e >= E_EDGES) return;
    const int src = ei[e];
    const int tgt = ei[E_EDGES + e];
    const float ev    = leaky(s_t[(size_t)tgt * H_HEADS + hd] +
                              s_s[(size_t)src * H_HEADS + hd]);
    const float alpha = __expf(ev - segmax[(size_t)tgt * H_HEADS + hd]) /
                        segsum[(size_t)tgt * H_HEADS + hd];
    const float val = alpha * h[(size_t)src * F_TOT + f];
    atomicAdd(&acc[(size_t)tgt * F_TOT + f], val);
}

// ---------------------------------------------------------------------------
// Kernel 7: out = elu(acc + deg * h)   (skip connection: sum_edges h_i = deg*h_i)
// ---------------------------------------------------------------------------
__global__ void gat_finalize(const float* __restrict__ acc,
                             const float* __restrict__ h,
                             const float* __restrict__ deg,
                             float* __restrict__ out)
{
    int idx = blockIdx.x * blockDim.x + threadIdx.x;
    if (idx >= N_NODES * F_TOT) return;
    const int n = idx >> 7;
    const float x = acc[idx] + deg[n] * h[idx];
    out[idx] = x > 0.f ? x : expm1f(x);
}

// ---------------------------------------------------------------------------
extern "C" void kernel_launch(void* const* d_in, const int* in_sizes, int n_in,
                              void* d_out, int out_size, void* d_ws, size_t ws_size,
                              hipStream_t stream)
{
    const float* X    = (const float*)d_in[0];  // [N, 128]
    const int*   ei   = (const int*)  d_in[1];  // [2, E]
    const float* W    = (const float*)d_in[2];  // [128, 128]
    const float* bias = (const float*)d_in[3];  // [128]
    const float* a    = (const float*)d_in[4];  // [32]
    float* out = (float*)d_out;

    // Workspace carve-up (~58 MB total)
    float* ws     = (float*)d_ws;
    float* h      = ws;                                   // N*128
    float* acc    = h      + (size_t)N_NODES * F_TOT;     // N*128
    float* s_t    = acc    + (size_t)N_NODES * F_TOT;     // N*8
    float* s_s    = s_t    + (size_t)N_NODES * H_HEADS;   // N*8
    float* segmax = s_s    + (size_t)N_NODES * H_HEADS;   // N*8
    float* segsum = segmax + (size_t)N_NODES * H_HEADS;   // N*8
    float* deg    = segsum + (size_t)N_NODES * H_HEADS;   // N

    const int nf = N_NODES * F_TOT;

    gat_project_wmma<<<N_NODES / 16, 256, 0, stream>>>(X, W, bias, h);
    gat_init<<<(nf + 255) / 256, 256, 0, stream>>>(acc, segmax, segsum, deg);
    gat_node_scores<<<(N_NODES * H_HEADS + 255) / 256, 256, 0, stream>>>(h, a, s_t, s_s);
    gat_edge_max<<<(E_EDGES + 255) / 256, 256, 0, stream>>>(ei, s_t, s_s, segmax, deg);
    gat_edge_sum<<<(E_EDGES + 255) / 256, 256, 0, stream>>>(ei, s_t, s_s, segmax, segsum);
    gat_edge_scatter<<<(E_EDGES + 1) / 2, 256, 0, stream>>>(ei, h, s_t, s_s, segmax, segsum, acc);
    gat_finalize<<<(nf + 255) / 256, 256, 0, stream>>>(acc, h, deg, out);
}